// YOLOv2Loss_23854248362093
// MI455X (gfx1250) — compile-verified
//
#include <hip/hip_runtime.h>
#include <hip/hip_bf16.h>
#include <stdint.h>

// Problem constants (match the reference)
#define BSZ 32
#define NA  5
#define NY  52
#define NX  52
#define NC  80
#define MGT 50
#define SG  (NY * NX)        // 2704 cells
#define CH  (5 + NC)         // 85 channels in preds
#define OC  10               // iou_t, iou_m, box_t(4), box_m, box_s, cls_t, cls_m
#define POS (NA * SG)        // 13520 (a, cell) positions per batch

__device__ __forceinline__ float sigmoidf_(float x) {
    return 1.0f / (1.0f + __expf(-x));
}

// ---------------------------------------------------------------------------
// Phase A: dense max-IoU + output init.
// Grid: (ceil(POS/256), BSZ). Block: 256 threads (8 wave32s).
// GT boxes + valid flags for the batch are staged into LDS with CDNA5 async
// loads (ASYNCcnt path); the 50-deep IoU loop then runs on ds_load_b128 pairs.
// ---------------------------------------------------------------------------
__global__ __launch_bounds__(256) void yolo_phaseA(
    const float* __restrict__ preds,
    const float* __restrict__ gt_boxes,
    const int*   __restrict__ gt_valid,
    const float* __restrict__ anchors,
    float*       __restrict__ out,
    float*       __restrict__ max_iou_ws)
{
    __shared__ __align__(16) float4 gbox[MGT];   // async target; converted in place
    __shared__ __align__(16) float4 gmeta[MGT];  // x = area, y = valid flag
    __shared__ __align__(16) int    svalid[MGT]; // async target for gt_valid
    __shared__ int nobj;

    const int b   = blockIdx.y;
    const int tid = threadIdx.x;
    const int p   = blockIdx.x * blockDim.x + tid;
    const bool act = (p < POS);
    const int pc  = act ? p : 0;
    const int a   = pc / SG;
    const int s   = pc - a * SG;

    const float* pp = preds + ((size_t)(b * NA + a) * SG + s) * CH;
    // CDNA5 WGP-scope prefetch (TH=0 => fill all cache levels): warm the
    // strided preds row while the async LDS staging + barrier hides latency.
    asm volatile("global_prefetch_b8 %0, off" :: "v"(pp) : "memory");

    // CDNA5 async tensor path: stage 50 GT boxes (16B/lane) + valid flags
    // (4B/lane) into LDS; tracked on ASYNCcnt.
    if (tid < MGT) {
        unsigned boxoff = (unsigned)(uintptr_t)(&gbox[tid]);
        const float* gsrc = gt_boxes + ((size_t)b * MGT + tid) * 4;
        asm volatile("global_load_async_to_lds_b128 %0, %1, off"
                     :: "v"(boxoff), "v"(gsrc) : "memory");
        unsigned valoff = (unsigned)(uintptr_t)(&svalid[tid]);
        const int* vsrc = gt_valid + (size_t)b * MGT + tid;
        asm volatile("global_load_async_to_lds_b32 %0, %1, off"
                     :: "v"(valoff), "v"(vsrc) : "memory");
    }
    asm volatile("s_wait_asynccnt 0" ::: "memory");
    __syncthreads();

    if (tid < MGT) {
        float4 r = gbox[tid];
        float cx = r.x * (float)NX;
        float cy = r.y * (float)NY;
        float w  = r.z * (float)NX;
        float h  = r.w * (float)NY;
        gbox[tid]  = make_float4(cx - 0.5f * w, cy - 0.5f * h,
                                 cx + 0.5f * w, cy + 0.5f * h);
        gmeta[tid] = make_float4(w * h, (svalid[tid] > 0) ? 1.0f : 0.0f,
                                 0.0f, 0.0f);
    }
    __syncthreads();
    if (tid == 0) {
        int n = 0;
        for (int m = 0; m < MGT; ++m) n += (svalid[m] > 0);
        nobj = n;
    }
    __syncthreads();

    if (!act) return;

    const int gy = s / NX;
    const int gx = s - gy * NX;

    const float t0 = pp[0], t1 = pp[1], t2 = pp[2], t3 = pp[3];
    const float aw = anchors[2 * a], ah = anchors[2 * a + 1];
    const float px = sigmoidf_(t0) + (float)gx;
    const float py = sigmoidf_(t1) + (float)gy;
    const float pw = __expf(t2) * aw;
    const float ph = __expf(t3) * ah;
    const float ax1 = px - 0.5f * pw, ay1 = py - 0.5f * ph;
    const float ax2 = px + 0.5f * pw, ay2 = py + 0.5f * ph;
    const float parea = pw * ph;

    float maxi = -1.0f;
    #pragma unroll 5
    for (int m = 0; m < MGT; ++m) {
        float4 gb = gbox[m];    // ds_load_b128
        float4 gm = gmeta[m];   // ds_load_b128
        float ltx = fmaxf(ax1, gb.x);
        float lty = fmaxf(ay1, gb.y);
        float rbx = fminf(ax2, gb.z);
        float rby = fminf(ay2, gb.w);
        float iw = fmaxf(rbx - ltx, 0.0f);
        float ih = fmaxf(rby - lty, 0.0f);
        float inter = iw * ih;
        float iou = inter / (parea + gm.x - inter + 1e-9f);
        iou = (gm.y > 0.5f) ? iou : -1.0f;
        maxi = fmaxf(maxi, iou);
    }

    const size_t lin = (size_t)(b * NA + a) * SG + s;
    max_iou_ws[lin] = maxi;

    float imask = 1.0f;
    if (maxi >= 0.5f) imask = 0.0f;
    if (nobj == 0)    imask = 0.0f;

    // Row stride is 40 B => every float2 slot is 8B aligned: 5x b64 stores.
    float2* o2 = (float2*)(out + lin * OC);
    o2[0] = make_float2(0.0f, imask);  // iou_target, iou_mask
    o2[1] = make_float2(0.0f, 0.0f);   // box_target xy
    o2[2] = make_float2(0.0f, 0.0f);   // box_target wh
    o2[3] = make_float2(0.0f, 0.0f);   // box_mask, box_scale
    o2[4] = make_float2(0.0f, 0.0f);   // class_target, class_mask
}

// ---------------------------------------------------------------------------
// Phase B: per-GT anchor assignment + scatter. One wave32, one lane per batch;
// serial over m within a batch so duplicate cells resolve last-write-wins
// deterministically (matches in-order scatter-set).
// ---------------------------------------------------------------------------
__global__ __launch_bounds__(32) void yolo_phaseB(
    const float* __restrict__ preds,
    const float* __restrict__ gt_boxes,
    const int*   __restrict__ gt_cls,
    const int*   __restrict__ gt_valid,
    const float* __restrict__ anchors,
    const float* __restrict__ max_iou_ws,
    float*       __restrict__ out)
{
    const int b = threadIdx.x;
    if (b >= BSZ) return;

    float aw[NA], ah[NA];
    #pragma unroll
    for (int a = 0; a < NA; ++a) {
        aw[a] = anchors[2 * a];
        ah[a] = anchors[2 * a + 1];
    }

    for (int m = 0; m < MGT; ++m) {
        if (gt_valid[b * MGT + m] <= 0) continue;

        const float* gb = gt_boxes + ((size_t)b * MGT + m) * 4;
        const float gx = gb[0] * (float)NX;
        const float gy = gb[1] * (float)NY;
        const float gw = gb[2] * (float)NX;
        const float gh = gb[3] * (float)NY;

        int cix = (int)floorf(gx); cix = min(max(cix, 0), NX - 1);
        int ciy = (int)floorf(gy); ciy = min(max(ciy, 0), NY - 1);
        const int cell = ciy * NX + cix;

        const float g1 = gx - 0.5f * gw, g2 = gy - 0.5f * gh;
        const float g3 = gx + 0.5f * gw, g4 = gy + 0.5f * gh;
        const float gar = gw * gh;
        const float acx = (float)cix, acy = (float)ciy;

        int astar = 0;
        float best = -1e30f;
        #pragma unroll
        for (int a = 0; a < NA; ++a) {
            float a1 = acx - 0.5f * aw[a], a2 = acy - 0.5f * ah[a];
            float a3 = acx + 0.5f * aw[a], a4 = acy + 0.5f * ah[a];
            float ltx = fmaxf(a1, g1), lty = fmaxf(a2, g2);
            float rbx = fminf(a3, g3), rby = fminf(a4, g4);
            float iw = fmaxf(rbx - ltx, 0.0f);
            float ih = fmaxf(rby - lty, 0.0f);
            float inter = iw * ih;
            float ov = inter / (aw[a] * ah[a] + gar - inter + 1e-9f);
            if (ov > best) { best = ov; astar = a; }   // first max wins (argmax)
        }

        const float d0 = gx - acx;
        const float d1 = gy - acy;
        const float d2 = gw / aw[astar];
        const float d3 = gh / ah[astar];

        const size_t lin = (size_t)(b * NA + astar) * SG + cell;
        const float* pp = preds + lin * CH;
        const float pw = __expf(pp[2]) * aw[astar];
        const float ph = __expf(pp[3]) * ah[astar];
        const float bscale = 2.0f - (pw / (float)NX) * (ph / (float)NY);
        const float mi = max_iou_ws[lin];

        float2* o2 = (float2*)(out + lin * OC);
        o2[0] = make_float2(mi, 2.0f);                        // iou_target, iou_mask
        o2[1] = make_float2(d0, d1);                          // box_target xy
        o2[2] = make_float2(d2, d3);                          // box_target wh
        o2[3] = make_float2(1.0f, bscale);                    // box_mask, box_scale
        o2[4] = make_float2((float)gt_cls[b * MGT + m], 1.0f);// class_target, class_mask
    }
}

extern "C" void kernel_launch(void* const* d_in, const int* in_sizes, int n_in,
                              void* d_out, int out_size, void* d_ws, size_t ws_size,
                              hipStream_t stream) {
    (void)in_sizes; (void)n_in; (void)out_size; (void)ws_size;

    const float* preds    = (const float*)d_in[0];
    const float* gt_boxes = (const float*)d_in[1];
    const int*   gt_cls   = (const int*)  d_in[2];
    const int*   gt_valid = (const int*)  d_in[3];
    const float* anchors  = (const float*)d_in[4];
    float*       out      = (float*)d_out;
    float*       max_iou  = (float*)d_ws;   // BSZ*NA*SG floats (~1.73 MB)

    dim3 blockA(256);
    dim3 gridA((POS + 255) / 256, BSZ);
    yolo_phaseA<<<gridA, blockA, 0, stream>>>(preds, gt_boxes, gt_valid,
                                              anchors, out, max_iou);

    yolo_phaseB<<<1, 32, 0, stream>>>(preds, gt_boxes, gt_cls, gt_valid,
                                      anchors, max_iou, out);
}